// SSRP_T_30992484008196
// MI455X (gfx1250) — compile-verified
//
#include <hip/hip_runtime.h>
#include <stdint.h>

// Problem constants (reference: x (16,128,128,1000) fp32, W=4, K=12)
constexpr int T_LEN = 1000;          // samples per row
constexpr int F_DIM = 128;           // rows per (b,c)
constexpr int TW    = T_LEN - 4 + 1; // 997 window outputs
constexpr int WAVES = 8;             // 256 threads / wave32
constexpr int ROWS_PER_WAVE = F_DIM / WAVES; // 16
constexpr int BUF_F = 1040;          // padded floats per LDS row buffer (>= 1004)

// Async global -> LDS DMA, 16B per active lane. INST_OFFSET applies to BOTH
// the global address and the LDS address (CDNA5 ISA), so one
// base VGPR pair + one LDS-offset VGPR covers the whole row.
// The input is a 1 GB single-use stream vs a 192 MB L2: mark loads
// non-temporal so the DMA doesn't thrash L2 (ISA §7.3 TH_LOAD_NT).
#define ASYNC_B128(LDSOFF, GADDR, OFFS)                                       \
  asm volatile("global_load_async_to_lds_b128 %0, %1, off offset:" #OFFS      \
               " th:TH_LOAD_NT"                                               \
               :: "v"(LDSOFF), "v"(GADDR) : "memory")

__device__ __forceinline__ void issue_row(uint32_t ldsoff, uint64_t gaddr, int lane) {
  // 7 full wave-wide chunks (7*512 = 3584 B) + 416 B tail on lanes 0..25.
  ASYNC_B128(ldsoff, gaddr, 0);
  ASYNC_B128(ldsoff, gaddr, 512);
  ASYNC_B128(ldsoff, gaddr, 1024);
  ASYNC_B128(ldsoff, gaddr, 1536);
  ASYNC_B128(ldsoff, gaddr, 2048);
  ASYNC_B128(ldsoff, gaddr, 2560);
  ASYNC_B128(ldsoff, gaddr, 3072);
  if (lane < 26) { ASYNC_B128(ldsoff, gaddr, 3584); }
}

// Sorted (ascending) top-12 kept in registers; branchless bubble insert.
__device__ __forceinline__ void insert12(float (&tk)[12], float v) {
  if (v > tk[0]) {
    float xx = v;
#pragma unroll
    for (int i = 0; i < 11; ++i) {
      const float lo = fminf(tk[i + 1], xx);
      xx = fmaxf(tk[i + 1], xx);
      tk[i] = lo;
    }
    tk[11] = xx;
  }
}

__global__ __launch_bounds__(256) void ssrp_topk_kernel(const float* __restrict__ x,
                                                        float* __restrict__ out) {
  __shared__ float smem[WAVES][2][BUF_F];
  __shared__ float wsum[WAVES];

  const int lane = threadIdx.x & 31;
  const int wave = threadIdx.x >> 5;
  const int bc   = blockIdx.x; // b*128 + c, matches flat output index

  const uint64_t row_bytes = (uint64_t)T_LEN * 4u;
  const uint64_t xbase = (uint64_t)(uintptr_t)x +
                         (uint64_t)bc * (uint64_t)F_DIM * row_bytes +
                         (uint64_t)lane * 16u;

  // Low 32 bits of a flat pointer into LDS == byte offset within the
  // workgroup LDS allocation (aperture translation truncates to addr[31:0]).
  const uint32_t lds0 = (uint32_t)(uintptr_t)(&smem[wave][0][0]) + (uint32_t)(lane * 16);
  const uint32_t lds1 = (uint32_t)(uintptr_t)(&smem[wave][1][0]) + (uint32_t)(lane * 16);

  float facc = 0.0f;

  // Preload row r=0 (f = wave) into buffer 0.
  issue_row(lds0, xbase + (uint64_t)wave * row_bytes, lane);

#pragma unroll 1
  for (int r = 0; r < ROWS_PER_WAVE; ++r) {
    const int pb = r & 1;

    if (r + 1 < ROWS_PER_WAVE) {
      const int nf = wave + WAVES * (r + 1);
      // Make sure our previous ds reads of the target buffer have retired
      // before the DMA overwrites it, then prefetch next row.
      asm volatile("s_wait_dscnt 0" ::: "memory");
      issue_row(pb ? lds0 : lds1, xbase + (uint64_t)nf * row_bytes, lane);
      // 16 async ops outstanding; <=8 left means current buffer is complete
      // (async loads complete in order).
      asm volatile("s_wait_asynccnt 8" ::: "memory");
    } else {
      asm volatile("s_wait_asynccnt 0" ::: "memory");
    }

    const float* buf = &smem[wave][pb][0];

    float tk[12];
#pragma unroll
    for (int i = 0; i < 12; ++i) tk[i] = -__builtin_huge_valf();

    // Lane handles 4 consecutive windows per chunk: t0 = i*128 + lane*4.
#pragma unroll
    for (int i = 0; i < 8; ++i) {
      const int t0 = i * 128 + lane * 4;
      if (t0 < TW) {
        const float4 h = *(const float4*)(buf + t0);      // x[t0..t0+3]
        const float4 g = *(const float4*)(buf + t0 + 4);  // x[t0+4..t0+7]
        const float s0 = h.x + h.y + h.z + h.w;           // raw window sums
        const float s1 = s0 - h.x + g.x;
        const float s2 = s1 - h.y + g.y;
        const float s3 = s2 - h.z + g.z;
        insert12(tk, s0);
        if (t0 + 1 < TW) insert12(tk, s1);
        if (t0 + 2 < TW) insert12(tk, s2);
        if (t0 + 3 < TW) insert12(tk, s3);
      }
    }

    // Wave-wide merge: 12 rounds of max-reduce; owner pops its candidate.
    float rowsum = 0.0f;
#pragma unroll 1
    for (int k = 0; k < 12; ++k) {
      const float c = tk[11];
      float m = c;
#pragma unroll
      for (int off = 16; off >= 1; off >>= 1)
        m = fmaxf(m, __shfl_xor(m, off, 32));
      rowsum += m;
      const unsigned bal = (unsigned)__ballot(c == m);
      if (lane == __ffs(bal) - 1) {
#pragma unroll
        for (int i = 11; i >= 1; --i) tk[i] = tk[i - 1];
      }
    }
    facc += rowsum; // identical across lanes
  }

  if (lane == 0) wsum[wave] = facc;
  __syncthreads();
  if (threadIdx.x == 0) {
    float s = 0.0f;
#pragma unroll
    for (int w = 0; w < WAVES; ++w) s += wsum[w];
    // window sums -> window means (/4), top-k mean (/12), F mean (/128)
    out[bc] = s * (1.0f / (4.0f * 12.0f * 128.0f));
  }
}

extern "C" void kernel_launch(void* const* d_in, const int* in_sizes, int n_in,
                              void* d_out, int out_size, void* d_ws, size_t ws_size,
                              hipStream_t stream) {
  (void)in_sizes; (void)n_in; (void)d_ws; (void)ws_size;
  const float* x = (const float*)d_in[0];
  float* out = (float*)d_out;
  // out_size == 16*128 == 2048 blocks, one per (b,c)
  ssrp_topk_kernel<<<dim3((unsigned)out_size), dim3(256), 0, stream>>>(x, out);
}